// Features2FeaturesResidual_90202903151305
// MI455X (gfx1250) — compile-verified
//
#include <hip/hip_runtime.h>

// ---------------------------------------------------------------------------
// GCN-like 3-layer network for MI455X (gfx1250, wave32).
// GEMMs via V_WMMA_F32_16X16X4_F32 (fp32 matrix path, reference precision).
// Scatter/BN phases are L2-resident atomics (h1 + acc = 102MB < 192MB L2).
// ---------------------------------------------------------------------------

constexpr int   FD  = 128;   // feature dim
constexpr int   LST = 132;   // padded LDS row stride (floats): (4m+k)%64 -> no bank conflicts
constexpr float FEPS = 1e-5f;

typedef float v2f __attribute__((ext_vector_type(2)));
typedef float v8f __attribute__((ext_vector_type(8)));

// ---------------------------------------------------------------------------
// Helpers
// ---------------------------------------------------------------------------
__global__ __launch_bounds__(256) void zero_kernel(float* __restrict__ p, int n) {
  int i = blockIdx.x * 256 + threadIdx.x;
  if (i < n) p[i] = 0.0f;
}

__global__ __launch_bounds__(256) void count_kernel(const int* __restrict__ edges,
                                                    float* __restrict__ cnt, int E) {
  int e = blockIdx.x * 256 + threadIdx.x;
  if (e < E) {
    atomicAdd(&cnt[edges[2 * e + 0]], 1.0f);
    atomicAdd(&cnt[edges[2 * e + 1]], 1.0f);
  }
}

__global__ __launch_bounds__(256) void dinv_kernel(float* __restrict__ c, int n) {
  int i = blockIdx.x * 256 + threadIdx.x;
  if (i < n) c[i] = 1.0f / (1.0f + c[i]);
}

// ---------------------------------------------------------------------------
// Dual GEMM:  Y0 = X @ W0^T + b0  -> acc ;  Y1 = X @ W1^T + b1 -> h1
// One workgroup = 128 rows of X. 8 waves, wave w owns rows [16w,16w+16).
// ---------------------------------------------------------------------------
__device__ __forceinline__ void fill_tile(float* __restrict__ dst,
                                          const float* __restrict__ src,
                                          int rowbase, int nrows) {
  const int tid = threadIdx.x;
#pragma unroll
  for (int i = 0; i < 16; ++i) {
    int idx = tid + i * 256;        // 0..4095
    int r   = idx >> 5;             // row 0..127
    int c4  = (idx & 31) << 2;      // col 0..124 step 4
    float4 f = make_float4(0.f, 0.f, 0.f, 0.f);
    int v = rowbase + r;
    if (v < nrows) f = *(const float4*)(src + (size_t)v * FD + c4);
    *(float4*)(dst + r * LST + c4) = f;
  }
}

__device__ __forceinline__ void wmma_block(const float* __restrict__ Xs,
                                           const float* __restrict__ Ws,
                                           const float* __restrict__ bias,
                                           float* __restrict__ Y,
                                           int rowbase, int nrows) {
  const int tid  = threadIdx.x;
  const int wave = tid >> 5;
  const int lane = tid & 31;
  const int m0   = wave * 16;
  const int ml   = lane & 15;       // A: row M  / B: col N
  const int hi   = lane >> 4;       // half-wave select
  const int kh   = hi * 2;          // A/B f32 layout: vgpr j + half-wave -> K = 2*hi + j

  for (int nt = 0; nt < 8; ++nt) {
    const int n0 = nt * 16;
    v8f c = {0.f, 0.f, 0.f, 0.f, 0.f, 0.f, 0.f, 0.f};
#pragma unroll
    for (int kb = 0; kb < FD; kb += 4) {
      v2f a = *(const v2f*)(Xs + (m0 + ml) * LST + kb + kh);  // A[m, k..k+1]
      v2f b = *(const v2f*)(Ws + (n0 + ml) * LST + kb + kh);  // B[k, n] = W[n, k]
      c = __builtin_amdgcn_wmma_f32_16x16x4_f32(false, a, false, b,
                                                (short)0, c, false, false);
    }
    const int   n  = n0 + ml;
    const float bs = bias[n];
#pragma unroll
    for (int r = 0; r < 8; ++r) {   // C/D: vgpr r -> M = r + 8*hi
      const int v = rowbase + m0 + r + 8 * hi;
      if (v < nrows) Y[(size_t)v * FD + n] = c[r] + bs;
    }
  }
}

__global__ __launch_bounds__(256)
void gemm_dual_kernel(const float* __restrict__ X,
                      const float* __restrict__ W0l, const float* __restrict__ b0l,
                      const float* __restrict__ W1l, const float* __restrict__ b1l,
                      float* __restrict__ Y0, float* __restrict__ Y1, int nrows) {
  __shared__ float Xs[FD * LST];   // 67.6 KB
  __shared__ float Ws[FD * LST];   // 67.6 KB  (total 135 KB < WGP LDS)
  const int rowbase = blockIdx.x * FD;

  fill_tile(Xs, X, rowbase, nrows);
  fill_tile(Ws, W0l, 0, FD);
  __syncthreads();
  wmma_block(Xs, Ws, b0l, Y0, rowbase, nrows);
  __syncthreads();
  fill_tile(Ws, W1l, 0, FD);
  __syncthreads();
  wmma_block(Xs, Ws, b1l, Y1, rowbase, nrows);
}

// ---------------------------------------------------------------------------
// Edge scatter: acc[src] += h1[dst]; acc[dst] += h1[src].  One wave per edge.
// ---------------------------------------------------------------------------
__global__ __launch_bounds__(256)
void scatter_kernel(const int* __restrict__ edges, const float* __restrict__ h1,
                    float* __restrict__ acc, int E) {
  const long long t = (long long)blockIdx.x * 256 + threadIdx.x;
  if (t >= (long long)E * 32) return;
  const int e  = (int)(t >> 5);
  const int c4 = ((int)t & 31) << 2;
  const int s  = edges[2 * e + 0];
  const int d  = edges[2 * e + 1];
  float4 hs = *(const float4*)(h1 + (size_t)s * FD + c4);
  float4 hd = *(const float4*)(h1 + (size_t)d * FD + c4);
  float* as = acc + (size_t)s * FD + c4;
  float* ad = acc + (size_t)d * FD + c4;
  atomicAdd(as + 0, hd.x); atomicAdd(as + 1, hd.y);
  atomicAdd(as + 2, hd.z); atomicAdd(as + 3, hd.w);
  atomicAdd(ad + 0, hs.x); atomicAdd(ad + 1, hs.y);
  atomicAdd(ad + 2, hs.z); atomicAdd(ad + 3, hs.w);
}

// ---------------------------------------------------------------------------
// h = dinv*acc (in place) + per-feature sum / sumsq partial reduction.
// Block = 64 rows x 128 cols; thread owns one column, 32 rows.
// ---------------------------------------------------------------------------
__global__ __launch_bounds__(256)
void combine_kernel(const float* __restrict__ dinv, float* __restrict__ acc,
                    float* __restrict__ sums, int nrows) {
  __shared__ float s1[256];
  __shared__ float s2[256];
  const int tid  = threadIdx.x;
  const int d    = tid & (FD - 1);
  const int par  = tid >> 7;           // 0 or 1: even/odd rows
  const int base = blockIdx.x * 64;
  float s = 0.f, q = 0.f;
  for (int i = 0; i < 32; ++i) {
    int v = base + par + i * 2;
    if (v < nrows) {
      float h = dinv[v] * acc[(size_t)v * FD + d];
      acc[(size_t)v * FD + d] = h;
      s += h;
      q += h * h;
    }
  }
  s1[tid] = s;
  s2[tid] = q;
  __syncthreads();
  if (tid < FD) {
    atomicAdd(&sums[d],      s1[tid] + s1[tid + 128]);
    atomicAdd(&sums[FD + d], s2[tid] + s2[tid + 128]);
  }
}

// ---------------------------------------------------------------------------
// BatchNorm + (optional residual) + ReLU.
// ---------------------------------------------------------------------------
__global__ __launch_bounds__(256)
void bnrelu_kernel(const float* __restrict__ h, const float* __restrict__ sums,
                   const float* __restrict__ gamma, const float* __restrict__ beta,
                   const float* __restrict__ resid, float* __restrict__ out, int nrows) {
  const long long t = (long long)blockIdx.x * 256 + threadIdx.x;
  if (t >= (long long)nrows * 32) return;
  const int v  = (int)(t >> 5);
  const int c4 = ((int)t & 31) << 2;
  const float invV = 1.0f / (float)nrows;
  float4 hv = *(const float4*)(h + (size_t)v * FD + c4);
  float4 rv = make_float4(0.f, 0.f, 0.f, 0.f);
  if (resid) rv = *(const float4*)(resid + (size_t)v * FD + c4);
  float hvv[4] = {hv.x, hv.y, hv.z, hv.w};
  float rvv[4] = {rv.x, rv.y, rv.z, rv.w};
  float o[4];
#pragma unroll
  for (int j = 0; j < 4; ++j) {
    int dd = c4 + j;
    float mu  = sums[dd] * invV;
    float var = sums[FD + dd] * invV - mu * mu;
    float y = gamma[dd] * (hvv[j] - mu) * rsqrtf(var + FEPS) + beta[dd] + rvv[j];
    o[j] = fmaxf(y, 0.0f);
  }
  *(float4*)(out + (size_t)v * FD + c4) = make_float4(o[0], o[1], o[2], o[3]);
}

// ---------------------------------------------------------------------------
// Launch: degree pass, then 3x (gemm -> scatter -> combine -> bn+relu).
// ---------------------------------------------------------------------------
extern "C" void kernel_launch(void* const* d_in, const int* in_sizes, int n_in,
                              void* d_out, int out_size, void* d_ws, size_t ws_size,
                              hipStream_t stream) {
  const float* features = (const float*)d_in[0];
  const int*   edges    = (const int*)d_in[1];
  const float* W0       = (const float*)d_in[2];
  const float* b0       = (const float*)d_in[3];
  const float* W1       = (const float*)d_in[4];
  const float* b1       = (const float*)d_in[5];
  const float* gamma    = (const float*)d_in[6];
  const float* beta     = (const float*)d_in[7];

  const int V = in_sizes[0] / FD;
  const int E = in_sizes[1] / 2;

  float* dinv = (float*)d_ws;                 // V
  float* sums = dinv + V;                     // 256 (sum | sumsq)
  float* h1   = sums + 256;                   // V*FD
  float* acc  = h1 + (size_t)V * FD;          // V*FD
  float* xA   = acc + (size_t)V * FD;         // V*FD
  float* out  = (float*)d_out;

  // degree -> dinv
  zero_kernel<<<(V + 255) / 256, 256, 0, stream>>>(dinv, V);
  count_kernel<<<(E + 255) / 256, 256, 0, stream>>>(edges, dinv, E);
  dinv_kernel<<<(V + 255) / 256, 256, 0, stream>>>(dinv, V);

  const int gemm_blocks = (V + FD - 1) / FD;
  const int sc_blocks   = (int)(((long long)E * 32 + 255) / 256);
  const int cb_blocks   = (V + 63) / 64;
  const int bn_blocks   = (int)(((long long)V * 32 + 255) / 256);

  for (int l = 0; l < 3; ++l) {
    const float* xin  = (l == 0) ? features : xA;
    float*       xout = (l == 2) ? out : xA;
    const float* res  = (l == 2) ? features : nullptr;

    gemm_dual_kernel<<<gemm_blocks, 256, 0, stream>>>(
        xin,
        W0 + (size_t)l * FD * FD, b0 + (size_t)l * FD,
        W1 + (size_t)l * FD * FD, b1 + (size_t)l * FD,
        acc, h1, V);
    scatter_kernel<<<sc_blocks, 256, 0, stream>>>(edges, h1, acc, E);
    zero_kernel<<<1, 256, 0, stream>>>(sums, 256);
    combine_kernel<<<cb_blocks, 256, 0, stream>>>(dinv, acc, sums, V);
    bnrelu_kernel<<<bn_blocks, 256, 0, stream>>>(
        acc, sums, gamma + (size_t)l * FD, beta + (size_t)l * FD, res, xout, V);
  }
}